// QBatchNorm2d_64347200028820
// MI455X (gfx1250) — compile-verified
//
#include <hip/hip_runtime.h>

typedef __attribute__((ext_vector_type(2))) float v2f;
typedef __attribute__((ext_vector_type(8))) float v8f;

#define QBN_B   32
#define QBN_C   64
#define QBN_HW  3136                    // 56*56
#define QBN_N   (QBN_B * QBN_HW)        // 100352
#define QBN_EPS 1e-5

// ---------------------------------------------------------------------------
// Kernel 1: per-(batch, channel) partial moments.
// Regular (RT) loads on purpose: x (103 MB) fits in the 192 MB global L2, so
// this pass warms L2 and the apply pass re-reads x from L2, not HBM.
// ---------------------------------------------------------------------------
__global__ __launch_bounds__(256) void qbn_reduce_kernel(
    const float* __restrict__ x, float* __restrict__ partial)
{
    const int b   = blockIdx.x;
    const int c   = blockIdx.y;
    const int tid = threadIdx.x;

    const size_t HW = QBN_HW;
    const size_t compStride = (size_t)QBN_C * HW;   // stride between components
    const float* base = x + ((size_t)b * 4 * QBN_C + c) * HW;

    const float4* q0 = (const float4*)(base + 0 * compStride);
    const float4* q1 = (const float4*)(base + 1 * compStride);
    const float4* q2 = (const float4*)(base + 2 * compStride);
    const float4* q3 = (const float4*)(base + 3 * compStride);

    float s0 = 0.f, s1 = 0.f, s2 = 0.f, s3 = 0.f;
    float p00 = 0.f, p01 = 0.f, p02 = 0.f, p03 = 0.f;
    float p11 = 0.f, p12 = 0.f, p13 = 0.f;
    float p22 = 0.f, p23 = 0.f, p33 = 0.f;

    for (int i = tid; i < QBN_HW / 4; i += 256) {
        float4 a = q0[i], bb = q1[i], cc = q2[i], dd = q3[i];
#define QBN_ACC(e)                                            \
        { float e0 = a.e, e1 = bb.e, e2 = cc.e, e3 = dd.e;    \
          s0 += e0; s1 += e1; s2 += e2; s3 += e3;             \
          p00 += e0 * e0; p01 += e0 * e1; p02 += e0 * e2;     \
          p03 += e0 * e3; p11 += e1 * e1; p12 += e1 * e2;     \
          p13 += e1 * e3; p22 += e2 * e2; p23 += e2 * e3;     \
          p33 += e3 * e3; }
        QBN_ACC(x) QBN_ACC(y) QBN_ACC(z) QBN_ACC(w)
#undef QBN_ACC
    }

    __shared__ float red[14][256];
    float vals[14] = { s0, s1, s2, s3, p00, p01, p02, p03,
                       p11, p12, p13, p22, p23, p33 };
    for (int k = 0; k < 14; ++k) red[k][tid] = vals[k];
    __syncthreads();
    for (int off = 128; off > 0; off >>= 1) {
        if (tid < off)
            for (int k = 0; k < 14; ++k) red[k][tid] += red[k][tid + off];
        __syncthreads();
    }
    if (tid == 0) {
        float* outp = partial + ((size_t)c * QBN_B + b) * 14;
        for (int k = 0; k < 14; ++k) outp[k] = red[k][0];
    }
}

// ---------------------------------------------------------------------------
// Kernel 2: per-channel solve (64 channels, 1 thread each; fp64 — free).
// Builds A = cov + eps*I, Cholesky A = L L^T, U = L^T, Uinv = (L^-1)^T,
// then T = W * Uinv and b_eff = bias - T * mean.  Writes 20 floats/channel.
// ---------------------------------------------------------------------------
__global__ void qbn_solve_kernel(const float* __restrict__ partial,
                                 const float* __restrict__ weight,  // [4,4,C]
                                 const float* __restrict__ bias,    // [4,C]
                                 float* __restrict__ stats)         // [C,20]
{
    const int c = threadIdx.x;
    if (c >= QBN_C) return;

    float s[14];
    for (int k = 0; k < 14; ++k) s[k] = 0.f;
    for (int b = 0; b < QBN_B; ++b) {
        const float* p = partial + ((size_t)c * QBN_B + b) * 14;
        for (int k = 0; k < 14; ++k) s[k] += p[k];
    }

    const double invN = 1.0 / (double)QBN_N;
    double mean[4];
    for (int q = 0; q < 4; ++q) mean[q] = (double)s[q] * invN;

    // A = E[x x^T] - mean mean^T + eps I
    double A[4][4];
    {
        int idx = 4;
        for (int i = 0; i < 4; ++i)
            for (int j = i; j < 4; ++j) {
                double v = (double)s[idx++] * invN - mean[i] * mean[j];
                if (i == j) v += QBN_EPS;
                A[i][j] = v; A[j][i] = v;
            }
    }

    // Cholesky: A = L L^T (lower)
    double L[4][4] = {};
    for (int i = 0; i < 4; ++i) {
        double d = A[i][i];
        for (int k = 0; k < i; ++k) d -= L[i][k] * L[i][k];
        L[i][i] = sqrt(d);
        for (int j = i + 1; j < 4; ++j) {
            double v = A[j][i];
            for (int k = 0; k < i; ++k) v -= L[j][k] * L[i][k];
            L[j][i] = v / L[i][i];
        }
    }

    // Li = L^-1 (lower triangular)
    double Li[4][4] = {};
    for (int i = 0; i < 4; ++i) {
        Li[i][i] = 1.0 / L[i][i];
        for (int j = 0; j < i; ++j) {
            double v = 0.0;
            for (int k = j; k < i; ++k) v += L[i][k] * Li[k][j];
            Li[i][j] = -v / L[i][i];
        }
    }

    // T[i][j] = sum_k W[i][k] * Uinv[k][j],  Uinv[k][j] = Li[j][k] (k <= j)
    double T[4][4];
    for (int i = 0; i < 4; ++i)
        for (int j = 0; j < 4; ++j) {
            double v = 0.0;
            for (int k = 0; k <= j; ++k)
                v += (double)weight[(i * 4 + k) * QBN_C + c] * Li[j][k];
            T[i][j] = v;
        }

    double be[4];
    for (int i = 0; i < 4; ++i) {
        double v = (double)bias[i * QBN_C + c];
        for (int j = 0; j < 4; ++j) v -= T[i][j] * mean[j];
        be[i] = v;
    }

    float* st = stats + c * 20;
    for (int i = 0; i < 4; ++i)
        for (int j = 0; j < 4; ++j) st[i * 4 + j] = (float)T[i][j];
    for (int i = 0; i < 4; ++i) st[16 + i] = (float)be[i];
}

// ---------------------------------------------------------------------------
// Kernel 3: apply out = T*x + b_eff via V_WMMA_F32_16X16X4_F32.
//   D(16x16) = A(16x4) * B(4x16):
//     A rows 0..3 = T (rows 4..15 zero), K = component j
//     B = 16 consecutive pixels (cols) x 4 components (K)
//     D[m][n], m<4: output component m at pixel n -> lanes 0..15, VGPRs 0..3
// Unrolled 2 chunks/iteration for MLP.  Loads are non-temporal (last use of
// x; it should be discarded from L2 after this read), stores are
// non-temporal (write-once output must not evict x / thrash L2).
// Grid = (B, C); 8 waves/block round-robin over 196 chunks of the HW plane.
// ---------------------------------------------------------------------------
__global__ __launch_bounds__(256) void qbn_apply_kernel(
    const float* __restrict__ x,
    const float* __restrict__ stats,
    float* __restrict__ out)
{
    const int b    = blockIdx.x;
    const int c    = blockIdx.y;
    const int tid  = threadIdx.x;
    const int lane = tid & 31;
    const int wave = tid >> 5;              // 0..7

    const size_t HW = QBN_HW;
    const size_t compStride = (size_t)QBN_C * HW;
    const float* in = x   + ((size_t)b * 4 * QBN_C + c) * HW;
    float*       op = out + ((size_t)b * 4 * QBN_C + c) * HW;
    const float* st = stats + c * 20;

    // A fragment (f32 16x4): lane l -> M = l%16; VGPR0/1 = K = (l/16)*2 + {0,1}
    const int m  = lane & 15;
    const int kb = (lane >> 4) << 1;        // 0 or 2
    v2f a;
    a.x = (m < 4) ? st[m * 4 + kb]     : 0.0f;
    a.y = (m < 4) ? st[m * 4 + kb + 1] : 0.0f;

    const float be0 = st[16], be1 = st[17], be2 = st[18], be3 = st[19];

    // B fragment (f32 4x16): lane l -> N = l%16; VGPR0/1 = K = (l/16)*2 + {0,1}
    const float* src0 = in + (size_t)kb * compStride;        // component kb
    const float* src1 = in + (size_t)(kb + 1) * compStride;  // component kb+1

    const int NCHUNK = QBN_HW / 16;         // 196 (exact)
    for (int chunk = wave; chunk < NCHUNK; chunk += 16) {
        const int  chunk2 = chunk + 8;
        const bool have2  = (chunk2 < NCHUNK);   // wave-uniform

        const int p1 = chunk * 16 + m;
        v2f bf1, bf2;
        bf1.x = __builtin_nontemporal_load(src0 + p1);
        bf1.y = __builtin_nontemporal_load(src1 + p1);
        bf2.x = 0.0f; bf2.y = 0.0f;
        if (have2) {
            const int p2 = chunk2 * 16 + m;
            bf2.x = __builtin_nontemporal_load(src0 + p2);
            bf2.y = __builtin_nontemporal_load(src1 + p2);
        }

        v8f acc1 = {};
        v8f acc2 = {};
        acc1 = __builtin_amdgcn_wmma_f32_16x16x4_f32(
            false, a, false, bf1, (short)0, acc1, false, false);
        // Runs on zeros in the tail case; keeps EXEC all-ones at every WMMA.
        acc2 = __builtin_amdgcn_wmma_f32_16x16x4_f32(
            false, a, false, bf2, (short)0, acc2, false, false);

        if (lane < 16) {
            const int pp = chunk * 16 + lane;
            __builtin_nontemporal_store(acc1[0] + be0, op + 0 * compStride + pp);
            __builtin_nontemporal_store(acc1[1] + be1, op + 1 * compStride + pp);
            __builtin_nontemporal_store(acc1[2] + be2, op + 2 * compStride + pp);
            __builtin_nontemporal_store(acc1[3] + be3, op + 3 * compStride + pp);
        }
        if (have2 && lane < 16) {
            const int pp = chunk2 * 16 + lane;
            __builtin_nontemporal_store(acc2[0] + be0, op + 0 * compStride + pp);
            __builtin_nontemporal_store(acc2[1] + be1, op + 1 * compStride + pp);
            __builtin_nontemporal_store(acc2[2] + be2, op + 2 * compStride + pp);
            __builtin_nontemporal_store(acc2[3] + be3, op + 3 * compStride + pp);
        }
    }
}

// ---------------------------------------------------------------------------
extern "C" void kernel_launch(void* const* d_in, const int* in_sizes, int n_in,
                              void* d_out, int out_size, void* d_ws, size_t ws_size,
                              hipStream_t stream) {
    const float* x      = (const float*)d_in[0];   // [32, 256, 56, 56]
    const float* weight = (const float*)d_in[1];   // [4, 4, 64]
    const float* bias   = (const float*)d_in[2];   // [4, 64]
    float*       out    = (float*)d_out;

    float* partial = (float*)d_ws;                           // 2048 * 14 floats
    float* stats   = partial + (size_t)QBN_B * QBN_C * 14;   // 64 * 20 floats

    dim3 grid(QBN_B, QBN_C);
    qbn_reduce_kernel<<<grid, 256, 0, stream>>>(x, partial);
    qbn_solve_kernel<<<1, 64, 0, stream>>>(partial, weight, bias, stats);
    qbn_apply_kernel<<<grid, 256, 0, stream>>>(x, stats, out);
}